// STAR_4844723110491
// MI455X (gfx1250) — compile-verified
//
#include <hip/hip_runtime.h>
#include <cstddef>

typedef __attribute__((ext_vector_type(2))) float v2f;
typedef __attribute__((ext_vector_type(8))) float v8f;

namespace {
constexpr int Bn  = 16384;
constexpr int Fn  = 20;
constexpr int Vn  = 100000;
constexpr int En  = 16;
constexpr int INn = 320;   // F*E
constexpr int H0n = 256;
constexpr int H1n = 128;
constexpr int Dn  = 4;
constexpr int HAn = 64;
constexpr float EPSc = 1e-5f;
constexpr int STAT_ROWS = 512;
}

// ---------------------------------------------------------------------------
// Wave-level 16x16 output tile GEMM: O = act(A[16xK] @ W[KxN] + bias), fp32
// WMMA f32 16x16x4 fragment layout (wave32):
//   A (16x4):  lanes 0-15 -> M=lane, K=(0,1) in (v0,v1); lanes 16-31 -> K=(2,3)
//   B (4x16):  lanes 0-15 -> N=lane, K=(0,1) in (v0,v1); lanes 16-31 -> K=(2,3)
//   C/D:       vgpr v: lanes 0-15 -> M=v, N=lane; lanes 16-31 -> M=v+8
// ---------------------------------------------------------------------------
__device__ __forceinline__ void gemm16(const float* __restrict__ A, int lda,
                                       const float* __restrict__ W, int ldw,
                                       const float* __restrict__ bias,
                                       float* __restrict__ O, int ldo,
                                       int n0, int K, int lane, bool relu_out)
{
    const int m  = lane & 15;
    const int kk = (lane >> 4) << 1;      // 0 or 2
    const int nc = n0 + (lane & 15);

    v8f acc;
    const float bv = bias[nc];
#pragma unroll
    for (int v = 0; v < 8; ++v) acc[v] = bv;

#pragma unroll 4
    for (int k0 = 0; k0 < K; k0 += 4) {
        v2f a, b;
        a.x = A[m * lda + k0 + kk];                    // ds_load_b64 (contiguous pair)
        a.y = A[m * lda + k0 + kk + 1];
        b.x = W[(size_t)(k0 + kk) * ldw + nc];         // L2-resident weight stream
        b.y = W[(size_t)(k0 + kk + 1) * ldw + nc];
        acc = __builtin_amdgcn_wmma_f32_16x16x4_f32(false, a, false, b,
                                                    (short)0, acc, false, false);
    }

    const int rb = (lane >> 4) << 3;
#pragma unroll
    for (int v = 0; v < 8; ++v) {
        float x = acc[v];
        if (relu_out) x = fmaxf(x, 0.0f);
        O[(rb + v) * ldo + nc] = x;
    }
}

// ---------------------------------------------------------------------------
// Kernel 1: embedding gather  x[b, f*16..f*16+15] = emb[f, features[b,f], :]
// ---------------------------------------------------------------------------
__global__ __launch_bounds__(256) void star_gather(const float* __restrict__ emb,
                                                   const int* __restrict__ feats,
                                                   float* __restrict__ xg)
{
    int i = blockIdx.x * 256 + threadIdx.x;      // one thread per (b, f)
    if (i >= Bn * Fn) return;
    int b = i / Fn, f = i % Fn;
    int idx = feats[i];
    const float4* src = (const float4*)(emb + ((size_t)f * Vn + (size_t)idx) * En);
    float4* dst = (float4*)(xg + (size_t)b * INn + f * En);
    dst[0] = src[0]; dst[1] = src[1]; dst[2] = src[2]; dst[3] = src[3];
}

// ---------------------------------------------------------------------------
// Kernel 2: per-domain segment sums of x and x^2 (+ counts)
// grid (IN/64, B/STAT_ROWS), block (64,4)
// ---------------------------------------------------------------------------
__global__ __launch_bounds__(256) void star_stats(const float* __restrict__ xg,
                                                  const int* __restrict__ dom,
                                                  float* __restrict__ sums,
                                                  float* __restrict__ sqs,
                                                  float* __restrict__ cnt)
{
    const int col  = blockIdx.x * 64 + threadIdx.x;
    const int rbeg = blockIdx.y * STAT_ROWS + threadIdx.y;
    const int rend = blockIdx.y * STAT_ROWS + STAT_ROWS;

    float s0 = 0.f, s1 = 0.f, s2 = 0.f, s3 = 0.f;
    float q0 = 0.f, q1 = 0.f, q2 = 0.f, q3 = 0.f;
    for (int r = rbeg; r < rend; r += 4) {
        float v  = xg[(size_t)r * INn + col];
        int   d  = dom[r];
        float v2 = v * v;
        if      (d == 0) { s0 += v; q0 += v2; }
        else if (d == 1) { s1 += v; q1 += v2; }
        else if (d == 2) { s2 += v; q2 += v2; }
        else             { s3 += v; q3 += v2; }
    }
    atomicAdd(&sums[0 * INn + col], s0); atomicAdd(&sqs[0 * INn + col], q0);
    atomicAdd(&sums[1 * INn + col], s1); atomicAdd(&sqs[1 * INn + col], q1);
    atomicAdd(&sums[2 * INn + col], s2); atomicAdd(&sqs[2 * INn + col], q2);
    atomicAdd(&sums[3 * INn + col], s3); atomicAdd(&sqs[3 * INn + col], q3);

    if (blockIdx.x == 0 && threadIdx.x == 0) {
        float c0 = 0.f, c1 = 0.f, c2 = 0.f, c3 = 0.f;
        for (int r = rbeg; r < rend; r += 4) {
            int d = dom[r];
            if (d == 0) c0 += 1.f; else if (d == 1) c1 += 1.f;
            else if (d == 2) c2 += 1.f; else c3 += 1.f;
        }
        atomicAdd(&cnt[0], c0); atomicAdd(&cnt[1], c1);
        atomicAdd(&cnt[2], c2); atomicAdd(&cnt[3], c3);
    }
}

// ---------------------------------------------------------------------------
// Kernel 3: fold BN into per-domain scale/shift
// ---------------------------------------------------------------------------
__global__ __launch_bounds__(256) void star_finalize(const float* __restrict__ sums,
                                                     const float* __restrict__ sqs,
                                                     const float* __restrict__ cnt,
                                                     const float* __restrict__ gamma,
                                                     const float* __restrict__ beta,
                                                     float* __restrict__ scale,
                                                     float* __restrict__ shift)
{
    int i = blockIdx.x * 256 + threadIdx.x;
    if (i >= Dn * INn) return;
    int d = i / INn;
    float c    = fmaxf(cnt[d], 1.0f);
    float mean = sums[i] / c;
    float var  = sqs[i] / c - mean * mean;
    float inv  = rsqrtf(var + EPSc);
    float sc   = gamma[i] * inv;
    scale[i] = sc;
    shift[i] = beta[i] - mean * sc;
}

// ---------------------------------------------------------------------------
// Kernel 4: fused per-16-row-tile compute (all GEMMs on WMMA f32)
// 128 threads = 4 wave32 per block; block handles rows [16*blockIdx.x, +16)
// ---------------------------------------------------------------------------
__global__ __launch_bounds__(128) void star_main(
    const float* __restrict__ xg,     const float* __restrict__ scale,
    const float* __restrict__ shift,  const int*   __restrict__ dom_id,
    const float* __restrict__ sw0, const float* __restrict__ sb0,
    const float* __restrict__ sw1, const float* __restrict__ sb1,
    const float* __restrict__ sw2, const float* __restrict__ sb2,
    const float* __restrict__ dw1, const float* __restrict__ db1,
    const float* __restrict__ dw2, const float* __restrict__ db2,
    const float* __restrict__ aw1, const float* __restrict__ ab1,
    const float* __restrict__ aw2, const float* __restrict__ ab2,
    float* __restrict__ out)
{
    __shared__ float xs[16][INn + 2];   // x tile, later BN'd in place (padded stride)
    __shared__ float s0[16][H0n + 2];   // shared tower hidden 0
    __shared__ float hb[16][H1n + 2];   // hidden-128 buffer (shared L1 / domain MLPs)
    __shared__ float ha[16][HAn + 2];   // aux hidden
    __shared__ float red[16][8];
    __shared__ float sharedv[16], dallv[16], auxv[16];
    __shared__ int   domv[16];

    const int tid  = threadIdx.x;
    const int lane = tid & 31;
    const int wid  = tid >> 5;
    const int r0   = blockIdx.x * 16;

    // ---- load x tile + domain ids
    for (int i = tid; i < 16 * INn; i += 128) {
        int r = i / INn, c = i % INn;
        xs[r][c] = xg[(size_t)(r0 + r) * INn + c];
    }
    if (tid < 16) domv[tid] = dom_id[r0 + tid];
    __syncthreads();                                                   // (A)

    // ---- shared tower L0: s0 = relu(x @ sw0 + sb0)   [16x256], K=320
    for (int t = 0; t < 4; ++t)
        gemm16(&xs[0][0], INn + 2, sw0, H0n, sb0, &s0[0][0], H0n + 2,
               (wid * 4 + t) * 16, INn, lane, true);
    // ---- aux L0: ha = relu(x @ aw1 + ab1)            [16x64], K=320
    gemm16(&xs[0][0], INn + 2, aw1, HAn, ab1, &ha[0][0], HAn + 2,
           wid * 16, INn, lane, true);
    __syncthreads();                                                   // (B)

    // ---- BN in place (xs readers are done): xn = x*scale[d] + shift[d]
    for (int i = tid; i < 16 * INn; i += 128) {
        int r = i / INn, c = i % INn;
        int d = domv[r];
        xs[r][c] = xs[r][c] * scale[d * INn + c] + shift[d * INn + c];
    }
    // ---- shared tower L1: hb = relu(s0 @ sw1 + sb1)  [16x128], K=256
    for (int t = 0; t < 2; ++t)
        gemm16(&s0[0][0], H0n + 2, sw1, H1n, sb1, &hb[0][0], H1n + 2,
               (wid * 2 + t) * 16, H0n, lane, true);
    __syncthreads();                                                   // (C)

    // ---- shared_out[m] = hb[m,:] . sw2 + sb2
    { int m = tid >> 3, seg = tid & 7; float p = 0.f;
      for (int j = 0; j < 16; ++j) p += hb[m][seg * 16 + j] * sw2[seg * 16 + j];
      red[m][seg] = p; }
    __syncthreads();
    if (tid < 16) { float s = sb2[0];
        for (int seg = 0; seg < 8; ++seg) s += red[tid][seg];
        sharedv[tid] = s; }
    __syncthreads();

    // ---- aux[m] = ha[m,:] . aw2 + ab2
    if (tid < 64) { int m = tid >> 2, seg = tid & 3; float p = 0.f;
        for (int j = 0; j < 16; ++j) p += ha[m][seg * 16 + j] * aw2[seg * 16 + j];
        red[m][seg] = p; }
    __syncthreads();
    if (tid < 16) { float s = ab2[0];
        for (int seg = 0; seg < 4; ++seg) s += red[tid][seg];
        auxv[tid] = s; }
    __syncthreads();

    // ---- domain MLPs: for each d, hb = relu(xn @ dw1[d] + db1[d]); route per row
    for (int d = 0; d < Dn; ++d) {
        for (int t = 0; t < 2; ++t)
            gemm16(&xs[0][0], INn + 2, dw1 + (size_t)d * INn * H1n, H1n,
                   db1 + d * H1n, &hb[0][0], H1n + 2,
                   (wid * 2 + t) * 16, INn, lane, true);
        __syncthreads();
        { int m = tid >> 3, seg = tid & 7; float p = 0.f;
          for (int j = 0; j < 16; ++j)
              p += hb[m][seg * 16 + j] * dw2[d * H1n + seg * 16 + j];
          red[m][seg] = p; }
        __syncthreads();
        if (tid < 16 && domv[tid] == d) { float s = db2[d];
            for (int seg = 0; seg < 8; ++seg) s += red[tid][seg];
            dallv[tid] = s; }
        __syncthreads();
    }

    if (tid < 16) out[r0 + tid] = sharedv[tid] * dallv[tid] + auxv[tid];
}

// ---------------------------------------------------------------------------
extern "C" void kernel_launch(void* const* d_in, const int* in_sizes, int n_in,
                              void* d_out, int out_size, void* d_ws, size_t ws_size,
                              hipStream_t stream)
{
    (void)in_sizes; (void)n_in; (void)out_size; (void)ws_size;

    const float* emb   = (const float*)d_in[0];
    const float* sw0   = (const float*)d_in[1];
    const float* sb0   = (const float*)d_in[2];
    const float* sw1   = (const float*)d_in[3];
    const float* sb1   = (const float*)d_in[4];
    const float* sw2   = (const float*)d_in[5];
    const float* sb2   = (const float*)d_in[6];
    const float* gamma = (const float*)d_in[7];
    const float* beta  = (const float*)d_in[8];
    const float* dw1   = (const float*)d_in[9];
    const float* db1   = (const float*)d_in[10];
    const float* dw2   = (const float*)d_in[11];
    const float* db2   = (const float*)d_in[12];
    const float* aw1   = (const float*)d_in[13];
    const float* ab1   = (const float*)d_in[14];
    const float* aw2   = (const float*)d_in[15];
    const float* ab2   = (const float*)d_in[16];
    const int*   feats = (const int*)d_in[17];
    const int*   dom   = (const int*)d_in[18];
    float* out = (float*)d_out;

    // workspace layout
    char*  ws     = (char*)d_ws;
    float* xg     = (float*)ws;
    size_t off    = (size_t)Bn * INn * sizeof(float);          // ~20 MB
    float* sums   = (float*)(ws + off);
    float* sqs    = sums + Dn * INn;
    float* cnt    = sqs + Dn * INn;
    float* scale  = cnt + 64;                                  // padded
    float* shift  = scale + Dn * INn;

    // zero the stats accumulators (atomics target) every call
    hipMemsetAsync(sums, 0, (size_t)(2 * Dn * INn + 64) * sizeof(float), stream);

    star_gather<<<(Bn * Fn + 255) / 256, 256, 0, stream>>>(emb, feats, xg);
    star_stats<<<dim3(INn / 64, Bn / STAT_ROWS), dim3(64, 4), 0, stream>>>(
        xg, dom, sums, sqs, cnt);
    star_finalize<<<(Dn * INn + 255) / 256, 256, 0, stream>>>(
        sums, sqs, cnt, gamma, beta, scale, shift);
    star_main<<<Bn / 16, 128, 0, stream>>>(
        xg, scale, shift, dom,
        sw0, sb0, sw1, sb1, sw2, sb2,
        dw1, db1, dw2, db2, aw1, ab1, aw2, ab2, out);
}